// QuanvolutionGen139_65481071404870
// MI455X (gfx1250) — compile-verified
//
#include <hip/hip_runtime.h>
#include <math.h>

// ---------------------------------------------------------------------------
// Quanvolution on MI455X (gfx1250, wave32).
//
// Algebraic fold: encoded state v = kron_w [cos(a_w/2), sin(a_w/2)] is REAL;
// the parameterized layers form a fixed 16x16 unitary U. Then
//   <Z_w> = v^T A_w v,  A_w = Re(U^H Z_w U)  (real symmetric 16x16).
// Kernel 1 builds the four A_w (4KB in d_ws). Kernel 2 does everything else
// with V_WMMA_F32_16X16X4_F32 tiles (fp32 WMMA -> no precision loss).
// ---------------------------------------------------------------------------

typedef float v2f __attribute__((ext_vector_type(2)));
typedef float v8f __attribute__((ext_vector_type(8)));

static __device__ __forceinline__ v8f wmma_f32_16x16x4(v2f a, v2f b, v8f c) {
  // (neg_a, A, neg_b, B, c_mod, C, reuse_a, reuse_b)
  return __builtin_amdgcn_wmma_f32_16x16x4_f32(false, a, false, b, (short)0, c,
                                               false, false);
}

// ------------------------- kernel 1: build A_w -----------------------------

static __device__ __forceinline__ float2 cmul(float2 a, float2 b) {
  return make_float2(a.x * b.x - a.y * b.y, a.x * b.y + a.y * b.x);
}
static __device__ __forceinline__ float2 cadd(float2 a, float2 b) {
  return make_float2(a.x + b.x, a.y + b.y);
}

// apply 2x2 gate on wire w (wire 0 = MSB of 4-bit index, matching the
// reference's (N,2,2,2,2) row-major flattening)
static __device__ __forceinline__ void apply1(float2* st, int w, float2 m00,
                                              float2 m01, float2 m10,
                                              float2 m11) {
  const int bit = 1 << (3 - w);
#pragma unroll
  for (int idx = 0; idx < 16; ++idx) {
    if (idx & bit) continue;
    float2 a = st[idx], c = st[idx | bit];
    st[idx]       = cadd(cmul(m00, a), cmul(m01, c));
    st[idx | bit] = cadd(cmul(m10, a), cmul(m11, c));
  }
}

static __device__ __forceinline__ void cnotg(float2* st, int cq, int tq) {
  const int cb = 1 << (3 - cq), tb = 1 << (3 - tq);
#pragma unroll
  for (int idx = 0; idx < 16; ++idx) {
    if ((idx & cb) && !(idx & tb)) {
      float2 tmp = st[idx];
      st[idx] = st[idx | tb];
      st[idx | tb] = tmp;
    }
  }
}

__global__ void build_A_kernel(const float* __restrict__ params,
                               float* __restrict__ A /* [4][16][16] */) {
  __shared__ float2 U[16][16];  // U[k][col]
  const int t = threadIdx.x;
  if (t < 16) {
    float2 st[16];
#pragma unroll
    for (int k = 0; k < 16; ++k) st[k] = make_float2(k == t ? 1.f : 0.f, 0.f);
#pragma unroll
    for (int d = 0; d < 2; ++d) {
#pragma unroll
      for (int w = 0; w < 4; ++w) {
        const float thx = params[(d * 4 + w) * 3 + 0];
        const float thy = params[(d * 4 + w) * 3 + 1];
        const float thz = params[(d * 4 + w) * 3 + 2];
        float cx, sx, cy, sy, cz, sz;
        sincosf(0.5f * thx, &sx, &cx);  // 24 params only: keep libm accuracy
        sincosf(0.5f * thy, &sy, &cy);
        sincosf(0.5f * thz, &sz, &cz);
        // RX: [[c, -i s], [-i s, c]]
        apply1(st, w, make_float2(cx, 0.f), make_float2(0.f, -sx),
               make_float2(0.f, -sx), make_float2(cx, 0.f));
        // RY: [[c, -s], [s, c]]
        apply1(st, w, make_float2(cy, 0.f), make_float2(-sy, 0.f),
               make_float2(sy, 0.f), make_float2(cy, 0.f));
        // RZ: diag(e^{-i th/2}, e^{+i th/2})
        apply1(st, w, make_float2(cz, -sz), make_float2(0.f, 0.f),
               make_float2(0.f, 0.f), make_float2(cz, sz));
      }
#pragma unroll
      for (int w = 0; w < 3; ++w) cnotg(st, w, w + 1);
      cnotg(st, 3, 0);
    }
#pragma unroll
    for (int k = 0; k < 16; ++k) U[k][t] = st[k];
  }
  __syncthreads();
  if (t < 16) {
#pragma unroll
    for (int w = 0; w < 4; ++w) {
      for (int j = 0; j < 16; ++j) {
        float s = 0.f;
        for (int k = 0; k < 16; ++k) {
          const float z = ((k >> (3 - w)) & 1) ? -1.f : 1.f;
          const float2 a = U[k][t], c = U[k][j];
          s += z * (a.x * c.x + a.y * c.y);
        }
        A[(w * 16 + t) * 16 + j] = s;
      }
    }
  }
}

// ---------------------- kernel 2: fused quanv + GEMM -----------------------
//
// 256 blocks x 512 threads (16 waves). Block b owns images [16b, 16b+16).
// Wave wv handles patches p = wv, wv+16, ... Each wave builds a 16x16 tile V
// (16 images x 16 state components), computes T_w = V*A_w with WMMA, reduces
// F[i,w] = sum_n T_w[i,n]*V[i,n], and accumulates logits with one more WMMA
// per patch (F 16x4  x  W_p 4x16, classes padded to 16 with zeros).
// Partial logit tiles are reduced in LDS; wave 0 does masked log-softmax.
//
// WMMA fp32 operand layouts (CDNA5 ISA):
//  A 16x4 : lane L holds row M=L&15, vgpr j = K = 2*(L>>4)+j
//  B 4x16 : lane L holds col N=L&15, vgpr j = K = 2*(L>>4)+j
//  C 16x16: lane L holds col N=L&15, vgpr r = row M = r + 8*(L>>4)

#define PCHUNK 13  // ceil(196/16)

__global__ __launch_bounds__(512, 1) void quanv_kernel(
    const float* __restrict__ x, const float* __restrict__ W,
    const float* __restrict__ bias, const float* __restrict__ A,
    float* __restrict__ out) {
  __shared__ float smem[14024];
  float* sW = smem;          // [10][788]   (pad 784->788 to kill bank conflicts)
  float* sV = smem + 7880;   // [16 waves][16][20] (pad 16->20, 16B-aligned rows)
  float* sF = smem + 13000;  // [16 waves][4][16]
  float* sC = sV;            // alias after main loop: [16 waves][16][16]

  const int tid = threadIdx.x;
  // force wave id onto the scalar unit: makes the patch index p, its guard and
  // the per-wave LDS bases provably wave-uniform (s_cmp/s_cbranch instead of
  // v_cmp + exec save/restore around the WMMAs)
  const int wv = __builtin_amdgcn_readfirstlane(tid >> 5);
  const int lane = tid & 31;
  const int i = lane & 15;   // M row (image) / N col (class or component)
  const int hi = lane >> 4;  // K-half selector
  const int imgBase = blockIdx.x * 16;

  // stage classifier weights into LDS (rows 10..15 implicitly zero-padded)
  for (int idx = tid; idx < 7840; idx += 512) {
    const int c = idx / 784, k = idx - c * 784;
    sW[c * 788 + k] = W[idx];
  }

  // preload the four A_w as WMMA B-operands (held in registers for all 196 p)
  v2f bA[4][4];
#pragma unroll
  for (int w = 0; w < 4; ++w) {
#pragma unroll
    for (int kk = 0; kk < 4; ++kk) {
      const int row = 4 * kk + 2 * hi;
      v2f t = {A[(w * 16 + row) * 16 + i], A[(w * 16 + row + 1) * 16 + i]};
      bA[w][kk] = t;
    }
  }

  v8f C = {0.f, 0.f, 0.f, 0.f, 0.f, 0.f, 0.f, 0.f};
  __syncthreads();

  float* sVw = sV + wv * 320;
  float* sFw = sF + wv * 64;

  for (int it = 0; it < PCHUNK; ++it) {
    const int p = it * 16 + wv;            // scalar
    const bool active = (p < 196);         // scalar guard (EXEC stays full)
    float v[16];
    if (active) {
      const int pr = p / 14, pc = p - pr * 14;       // scalar
      const int pofs = pr * 56 + pc * 2;             // scalar
      const long base = (long)(imgBase + i) * 784 + pofs;
      const float2 top = *(const float2*)(x + base);       // pixels (0,0),(0,1)
      const float2 bot = *(const float2*)(x + base + 28);  // pixels (1,0),(1,1)
      // angles are pixel values in [0,1): native v_sin/v_cos are exact enough
      // (|err| ~ 2^-21) and avoid the libm Payne-Hanek huge-arg clause.
      float c0, s0, c1, s1, c2, s2, c3, s3;
      __sincosf(0.5f * top.x, &s0, &c0);
      __sincosf(0.5f * top.y, &s1, &c1);
      __sincosf(0.5f * bot.x, &s2, &c2);
      __sincosf(0.5f * bot.y, &s3, &c3);
      // v = [c0,s0] (x) [c1,s1] (x) [c2,s2] (x) [c3,s3], wire0 = MSB
      const float q[4] = {c0 * c1, c0 * s1, s0 * c1, s0 * s1};
      const float r4[4] = {c2 * c3, c2 * s3, s2 * c3, s2 * s3};
#pragma unroll
      for (int idx = 0; idx < 16; ++idx) v[idx] = q[idx >> 2] * r4[idx & 3];
      if (hi == 0) {  // one copy of V per wave into LDS (for C-layout reads)
#pragma unroll
        for (int j = 0; j < 16; ++j) sVw[i * 20 + j] = v[j];
      }
    }
    __syncthreads();
    if (active) {
      // V in C-layout: lane holds column n=i of rows r+8*hi
      float Vc[8];
#pragma unroll
      for (int r = 0; r < 8; ++r) Vc[r] = sVw[(r + 8 * hi) * 20 + i];
      // V column-chunks as A-operands (local: this lane owns image i's v)
      v2f vA[4];
#pragma unroll
      for (int kk = 0; kk < 4; ++kk) {
        v2f t = {v[4 * kk + 2 * hi], v[4 * kk + 2 * hi + 1]};
        vA[kk] = t;
      }
#pragma unroll
      for (int w = 0; w < 4; ++w) {
        v8f T = {0.f, 0.f, 0.f, 0.f, 0.f, 0.f, 0.f, 0.f};
#pragma unroll
        for (int kk = 0; kk < 4; ++kk)
          T = wmma_f32_16x16x4(vA[kk], bA[w][kk], T);  // T_w = V * A_w
        // F[m,w] = sum_n T_w[m,n]*V[m,n]: per-row lane reduction over 16 lanes
        float part[8];
#pragma unroll
        for (int r = 0; r < 8; ++r) part[r] = T[r] * Vc[r];
#pragma unroll
        for (int m = 1; m <= 8; m <<= 1) {
#pragma unroll
          for (int r = 0; r < 8; ++r) part[r] += __shfl_xor(part[r], m, 32);
        }
        if (i == 0) {  // lanes 0 and 16 publish rows 0-7 / 8-15
#pragma unroll
          for (int r = 0; r < 8; ++r) sFw[w * 16 + r + 8 * hi] = part[r];
        }
      }
    }
    __syncthreads();
    if (active) {
      // logits += F (16x4) x W_p (4x16)
      v2f fA = {sFw[(2 * hi) * 16 + i], sFw[(2 * hi + 1) * 16 + i]};
      const float* wrow = sW + i * 788 + 4 * p + 2 * hi;
      v2f wB = {(i < 10) ? wrow[0] : 0.f, (i < 10) ? wrow[1] : 0.f};
      C = wmma_f32_16x16x4(fA, wB, C);
    }
  }

  // reduce the 16 partial logit tiles
  __syncthreads();
#pragma unroll
  for (int r = 0; r < 8; ++r) sC[wv * 256 + (r + 8 * hi) * 16 + i] = C[r];
  __syncthreads();
  if (wv == 0) {
#pragma unroll
    for (int r = 0; r < 8; ++r) {
      float s = 0.f;
      for (int w2 = 0; w2 < 16; ++w2) s += sC[w2 * 256 + (r + 8 * hi) * 16 + i];
      const float val = (i < 10) ? s + bias[i] : -1e30f;
      float mx = val;
#pragma unroll
      for (int m = 1; m <= 8; m <<= 1) mx = fmaxf(mx, __shfl_xor(mx, m, 32));
      const float e = (i < 10) ? expf(val - mx) : 0.f;
      float se = e;
#pragma unroll
      for (int m = 1; m <= 8; m <<= 1) se += __shfl_xor(se, m, 32);
      if (i < 10)
        out[(long)(imgBase + r + 8 * hi) * 10 + i] = val - mx - logf(se);
    }
  }
}

// ------------------------------- launcher ----------------------------------

extern "C" void kernel_launch(void* const* d_in, const int* in_sizes, int n_in,
                              void* d_out, int out_size, void* d_ws,
                              size_t ws_size, hipStream_t stream) {
  (void)in_sizes; (void)n_in; (void)out_size; (void)ws_size;
  const float* x      = (const float*)d_in[0];  // (4096,1,28,28)
  const float* params = (const float*)d_in[1];  // (2,4,3)
  const float* W      = (const float*)d_in[2];  // (10,784)
  const float* b      = (const float*)d_in[3];  // (10,)
  float* out = (float*)d_out;                   // (4096,10)
  float* A = (float*)d_ws;                      // 4*16*16 floats

  build_A_kernel<<<1, 32, 0, stream>>>(params, A);
  quanv_kernel<<<256, 512, 0, stream>>>(x, W, b, A, out);
}